// TensorContractionLayer_80461917323957
// MI455X (gfx1250) — compile-verified
//
#include <hip/hip_runtime.h>
#include <cstdint>

// ---------------- problem constants ----------------
#define KTOT  4096          // 64*64 contraction per GEMM
#define MTOT  4096          // BATCH(64) * mode(64) rows
#define NTOT  256           // RANK*RANK cols
#define CHAIN_ELEMS (KTOT * NTOT)   // 4 MB fp32 per skip

// ---------------- GEMM tiling ----------------
#define TR_BM 128
#define TR_BN 64
#define TR_KC 64
#define NCHUNK (KTOT / TR_KC)
#define APAD  68            // row = 64 data + 4 pad dwords -> 272 B (16B aligned)
#define BPAD  68
#define PAD_INTERVAL_CODE 5 // pad every 64 dwords
#define PAD_AMOUNT_CODE   3 // pad by 4 dwords

typedef float v2f __attribute__((ext_vector_type(2)));
typedef float v8f __attribute__((ext_vector_type(8)));
// TDM / async builtin arg types (GCC-style vectors to match builtin decls)
typedef int          gv4i __attribute__((vector_size(16)));
typedef int          gv8i __attribute__((vector_size(32)));
typedef unsigned int gv4u __attribute__((vector_size(16)));

#if defined(__gfx1250__) && __has_builtin(__builtin_amdgcn_global_load_async_to_lds_b32)
#define HAS_ASYNC_LDS 1
#else
#define HAS_ASYNC_LDS 0
#endif
#if HAS_ASYNC_LDS && __has_builtin(__builtin_amdgcn_global_load_async_to_lds_b128)
#define HAS_B128 1
#else
#define HAS_B128 0
#endif
#if defined(__gfx1250__) && __has_builtin(__builtin_amdgcn_tensor_load_to_lds) && \
    __has_builtin(__builtin_amdgcn_s_wait_tensorcnt)
#define HAS_TDM 1
#else
#define HAS_TDM 0
#endif

// per-wave async-instruction counts for one chunk issue (compile-time)
#define NA_FAST (HAS_B128 ? 8 : 32)                  // A chunk, contiguous (skips 0,1)
#define NA_SLOW 32                                   // A chunk, strided (skip 2)
#define NB_CNT  (HAS_TDM ? 0 : (HAS_B128 ? 4 : 16))  // B chunk when not using TDM

#if HAS_ASYNC_LDS && __has_builtin(__builtin_amdgcn_s_wait_asynccnt)
#define WAIT_ASYNC(n) __builtin_amdgcn_s_wait_asynccnt(n)
#elif HAS_ASYNC_LDS
#define WAIT_ASYNC(n) asm volatile("s_wait_asynccnt 0x0" ::: "memory")  // conservative drain
#else
#define WAIT_ASYNC(n) do {} while (0)
#endif

#define AS1P32(p)  ((__attribute__((address_space(1))) int*)(uintptr_t)(p))
#define AS3P32(p)  ((__attribute__((address_space(3))) int*)(uint32_t)(uintptr_t)(p))
#define AS1P128(p) ((__attribute__((address_space(1))) gv4i*)(uintptr_t)(p))
#define AS3P128(p) ((__attribute__((address_space(3))) gv4i*)(uint32_t)(uintptr_t)(p))

__device__ __forceinline__ void cp_g2l_b32(float* lds, const float* g) {
#if HAS_ASYNC_LDS
  __builtin_amdgcn_global_load_async_to_lds_b32(AS1P32(g), AS3P32(lds), 0, 0);
#else
  *lds = *g;
#endif
}

__device__ __forceinline__ void cp_g2l_b128(float* lds, const float* g) {
#if HAS_B128
  __builtin_amdgcn_global_load_async_to_lds_b128(AS1P128(g), AS3P128(lds), 0, 0);
#elif HAS_ASYNC_LDS
  for (int j = 0; j < 4; ++j)
    __builtin_amdgcn_global_load_async_to_lds_b32(AS1P32(g + j), AS3P32(lds + j), 0, 0);
#else
  lds[0] = g[0]; lds[1] = g[1]; lds[2] = g[2]; lds[3] = g[3];
#endif
}

// ------------------------------------------------------------------
// Phase 1: build the three chain B-matrices in workspace (tiny FLOPs).
// chainB_s[k][a*16+c], k = kHi*64 + kLo:
//   skip0: sum_b core1[kHi][a][b] * core2[kLo][b][c]
//   skip1: sum_b core2[kLo][a][b] * core0[kHi][b][c]
//   skip2: sum_b core0[kHi][a][b] * core1[kLo][b][c]
// ------------------------------------------------------------------
__global__ __launch_bounds__(256) void chain_kernel(
    const float* __restrict__ c0, const float* __restrict__ c1,
    const float* __restrict__ c2, float* __restrict__ ws) {
  const int k = blockIdx.x;
  const int s = blockIdx.y;
  const int kHi = k >> 6, kLo = k & 63;

  const float* P; const float* Q; int p, q;
  if (s == 0)      { P = c1; p = kHi; Q = c2; q = kLo; }
  else if (s == 1) { P = c2; p = kLo; Q = c0; q = kHi; }
  else             { P = c0; p = kHi; Q = c1; q = kLo; }

  __shared__ float Ps[256];
  __shared__ float Qs[256];
  const int t = threadIdx.x;
  Ps[t] = P[p * 256 + t];
  Qs[t] = Q[q * 256 + t];
  __syncthreads();

  const int a = t >> 4, c = t & 15;
  float acc = 0.f;
#pragma unroll
  for (int b = 0; b < 16; ++b) acc += Ps[a * 16 + b] * Qs[b * 16 + c];

  ws[(size_t)s * CHAIN_ELEMS + (size_t)k * 256 + t] = acc;
}

// ------------------------------------------------------------------
// Phase 2: C_s(4096x256) = A_s(4096x4096 strided view of x) * B_s(4096x256)
// fp32 WMMA 16x16x4, 128x64 block tile, KC=64, double-buffered LDS,
// A via async-to-LDS, B via TDM tensor_load_to_lds (2D tile + LDS pad).
// ------------------------------------------------------------------
__global__ __launch_bounds__(256, 1) void tr_gemm(
    const float* __restrict__ X, const float* __restrict__ chains,
    float* __restrict__ out) {
  const int s = blockIdx.z;
  const int mBlock = blockIdx.x * TR_BM;
  const int nBlock = blockIdx.y * TR_BN;

  int SM, SH, SL;
  if (s == 0)      { SM = 4096; SH = 64;   SL = 1;  }
  else if (s == 1) { SM = 64;   SH = 4096; SL = 1;  }
  else             { SM = 1;    SH = 4096; SL = 64; }
  const bool fastA = (SL == 1);

  const float* __restrict__ Bmat = chains + (size_t)s * CHAIN_ELEMS;
  float* __restrict__ Cmat = out + (size_t)s * CHAIN_ELEMS;

  __shared__ __align__(16) float As[2][TR_BM][APAD];
  __shared__ __align__(16) float Bs[2][TR_KC][BPAD];

  const int tid  = threadIdx.x;
  const int lane = tid & 31;
  const int wave = tid >> 5;
  const int wm = (wave & 3) * 32;
  const int wn = (wave >> 2) * 32;
  const int lm = lane & 15;
  const int kh = (lane >> 4) << 1;

  v8f acc[2][2] = {};

  // ---- chunk issue helpers (lambdas keep uniform-branch structure) ----
  auto issueA = [&](int c, int buf) {
    const int kHiBase = (c * TR_KC >> 6) * SH;
    if (fastA) {
      // 64 contiguous floats per row: vectorized 16B copies
#pragma unroll 4
      for (int i = 0; i < 8; ++i) {
        const int flat = i * 256 + tid;          // 2048 vec4 slots
        const int row = flat >> 4;               // 16 vec4 per row
        const int c4  = (flat & 15) << 2;
        const int r = mBlock + row;
        const int gaddr = (r >> 6) * 262144 + (r & 63) * SM + kHiBase + c4;
        cp_g2l_b128(&As[buf][row][c4], X + gaddr);
      }
    } else {
      // skip2: memory-contiguous direction is the row index (SM==1)
#pragma unroll 4
      for (int i = 0; i < 32; ++i) {
        const int flat = i * 256 + tid;
        const int row = flat & 127;
        const int kk  = flat >> 7;
        const int r = mBlock + row;
        const int gaddr = (r >> 6) * 262144 + (r & 63) + kHiBase + kk * 64;
        cp_g2l_b32(&As[buf][row][kk], X + gaddr);
      }
    }
  };

  auto issueB = [&](int c, int buf) {
#if HAS_TDM
    if (wave == 0) {
      const uint64_t g = (uint64_t)(uintptr_t)(Bmat + (size_t)c * TR_KC * NTOT + nBlock);
      const uint32_t l = (uint32_t)(uintptr_t)&Bs[buf][0][0];
      gv4u g0; gv8i g1; gv4i g2; gv4i g3; gv8i g4;
      g2[0] = g2[1] = g2[2] = g2[3] = 0;
      g3[0] = g3[1] = g3[2] = g3[3] = 0;
      g4[0] = g4[1] = g4[2] = g4[3] = g4[4] = g4[5] = g4[6] = g4[7] = 0;
      g0[0] = 1u;                                   // count=1, normal mode
      g0[1] = l;                                    // lds_addr (bytes)
      g0[2] = (uint32_t)g;                          // global_addr[31:0]
      g0[3] = (uint32_t)(g >> 32) | (2u << 30);     // global_addr[56:32] | type=2
      const uint32_t td0 = 1u << 20, td1 = 1u << 20;  // generous tensor dims (no OOB)
      g1[0] = (int)((2u << 16) | (1u << 20) |       // data_size=4B | pad_enable
                    ((uint32_t)PAD_INTERVAL_CODE << 22) |
                    ((uint32_t)PAD_AMOUNT_CODE << 25));
      g1[1] = (int)((td0 & 0xFFFFu) << 16);         // tensor_dim0[15:0]
      g1[2] = (int)((td0 >> 16) | ((td1 & 0xFFFFu) << 16));
      g1[3] = (int)((td1 >> 16) | (64u << 16));     // tile_dim0 = 64
      g1[4] = 64;                                   // tile_dim1 = 64, tile_dim2 = 0
      g1[5] = NTOT;                                 // tensor_dim0_stride = 256
      g1[6] = 0;
      g1[7] = 0;
      __builtin_amdgcn_tensor_load_to_lds(g0, g1, g2, g3, g4, 0);
    }
#else
#pragma unroll
    for (int i = 0; i < 4; ++i) {
      const int flat = i * 256 + tid;               // 1024 vec4 slots
      const int row = flat >> 4;
      const int c4  = (flat & 15) << 2;
      cp_g2l_b128(&Bs[buf][row][c4],
                  Bmat + (size_t)(c * TR_KC + row) * NTOT + nBlock + c4);
    }
#endif
  };

  // ---- software pipeline: issue c+1, wait for c, compute c ----
  issueA(0, 0);
  issueB(0, 0);

  for (int c = 0; c < NCHUNK; ++c) {
    const int buf = c & 1;
    const bool more = (c + 1 < NCHUNK);
    if (more) { issueA(c + 1, buf ^ 1); issueB(c + 1, buf ^ 1); }

    if (more) { if (fastA) WAIT_ASYNC(NA_FAST + NB_CNT); else WAIT_ASYNC(NA_SLOW + NB_CNT); }
    else      { WAIT_ASYNC(0); }
#if HAS_TDM
    if (wave == 0) {
      if (more) __builtin_amdgcn_s_wait_tensorcnt(1);
      else      __builtin_amdgcn_s_wait_tensorcnt(0);
    }
#endif
    __syncthreads();

    const float (*A)[APAD] = As[buf];
    const float (*B)[BPAD] = Bs[buf];
#pragma unroll
    for (int k = 0; k < TR_KC; k += 4) {
      v2f a0 = *(const v2f*)&A[wm + lm][k + kh];
      v2f a1 = *(const v2f*)&A[wm + 16 + lm][k + kh];
      v2f b0, b1;
      b0.x = B[k + kh][wn + lm];
      b0.y = B[k + kh + 1][wn + lm];
      b1.x = B[k + kh][wn + 16 + lm];
      b1.y = B[k + kh + 1][wn + 16 + lm];

      acc[0][0] = __builtin_amdgcn_wmma_f32_16x16x4_f32(
          false, a0, false, b0, (short)0, acc[0][0], false, false);
      acc[0][1] = __builtin_amdgcn_wmma_f32_16x16x4_f32(
          false, a0, false, b1, (short)0, acc[0][1], false, false);
      acc[1][0] = __builtin_amdgcn_wmma_f32_16x16x4_f32(
          false, a1, false, b0, (short)0, acc[1][0], false, false);
      acc[1][1] = __builtin_amdgcn_wmma_f32_16x16x4_f32(
          false, a1, false, b1, (short)0, acc[1][1], false, false);
    }
    __syncthreads();
  }

  // ---- store C: VGPR v holds row (v + 8*(lane>=16)), col = lane&15 ----
  const int rsub = 8 * (lane >> 4);
#pragma unroll
  for (int mt = 0; mt < 2; ++mt) {
#pragma unroll
    for (int nt = 0; nt < 2; ++nt) {
      const int col = nBlock + wn + nt * 16 + lm;
      const int rbase = mBlock + wm + mt * 16 + rsub;
#pragma unroll
      for (int v = 0; v < 8; ++v)
        Cmat[(size_t)(rbase + v) * NTOT + col] = acc[mt][nt][v];
    }
  }
}

extern "C" void kernel_launch(void* const* d_in, const int* in_sizes, int n_in,
                              void* d_out, int out_size, void* d_ws, size_t ws_size,
                              hipStream_t stream) {
  const float* x  = (const float*)d_in[0];
  const float* c0 = (const float*)d_in[1];
  const float* c1 = (const float*)d_in[2];
  const float* c2 = (const float*)d_in[3];
  float* out = (float*)d_out;
  float* ws  = (float*)d_ws;   // needs 3 * 4 MB = 12 MB

  dim3 gridChain(KTOT, 3, 1);
  chain_kernel<<<gridChain, 256, 0, stream>>>(c0, c1, c2, ws);

  dim3 grid(MTOT / TR_BM, NTOT / TR_BN, 3);   // (32, 4, 3)
  tr_gemm<<<grid, 256, 0, stream>>>(x, ws, out);
}